// DataAwareGCN_17901423690367
// MI455X (gfx1250) — compile-verified
//
#include <hip/hip_runtime.h>
#include <hip/hip_bf16.h>

typedef float v2f __attribute__((ext_vector_type(2)));
typedef float v8f __attribute__((ext_vector_type(8)));

#define N_NODES 100000
#define N_EDGES 1600000
#define IN_DIM  256
#define HIDDEN  128
#define OUT_DIM 64

static inline int cdiv(long long a, long long b) { return (int)((a + b - 1) / b); }

// ---------------------------------------------------------------------------
// Utility kernels
// ---------------------------------------------------------------------------
__global__ void zero_i32_kernel(int* __restrict__ p, int n) {
    int i = blockIdx.x * blockDim.x + threadIdx.x;
    if (i < n) p[i] = 0;
}

__global__ void zero_f32_kernel(float* __restrict__ p, int n) {
    int i = blockIdx.x * blockDim.x + threadIdx.x;
    if (i < n) p[i] = 0.0f;
}

// one thread per edge: in-degree count over dst
__global__ void count_deg_kernel(const int* __restrict__ dst, int* __restrict__ cnt, int nE) {
    int e = blockIdx.x * blockDim.x + threadIdx.x;
    if (e < nE) atomicAdd(&cnt[dst[e]], 1);
}

// in-place: int count -> float rsqrt(count + 1)   (+1 = self loop)
__global__ void make_dinv_kernel(float* __restrict__ dinv_io, int n) {
    int i = blockIdx.x * blockDim.x + threadIdx.x;
    if (i < n) {
        int c = reinterpret_cast<const int*>(dinv_io)[i];
        dinv_io[i] = rsqrtf((float)c + 1.0f);
    }
}

// ---------------------------------------------------------------------------
// WMMA f32 GEMM:  C[M,N] = A[M,K] * B[K,N]   (row-major, exact f32 via
// V_WMMA_F32_16X16X4_F32). One wave computes a 16x16 tile; a block holds
// N/16 waves covering one 16-row M stripe. M = 100000 = 6250*16 (no tail).
//
// ISA fragment layouts (wave32):
//   A 16x4 f32 : lanes 0-15 -> M=lane, K={0,1} in v[0],v[1];
//                lanes 16-31 -> M=lane-16, K={2,3}
//   B 4x16 f32 : v[0]: lanes0-15 row K=0, lanes16-31 row K=2;
//                v[1]: lanes0-15 row K=1, lanes16-31 row K=3
//   C 16x16 f32: v[r]: lanes0-15 -> row r, lanes16-31 -> row r+8
// ---------------------------------------------------------------------------
template <int K, int N>
__global__ __launch_bounds__(32 * (N / 16))
void gemm_wmma_kernel(const float* __restrict__ A,
                      const float* __restrict__ B,
                      float* __restrict__ C) {
    const int lane  = threadIdx.x & 31;
    const int wave  = threadIdx.x >> 5;        // = N-tile index
    const int m16   = lane & 15;
    const int hi    = lane >> 4;               // 0 for lanes 0-15, 1 for 16-31
    const int row   = blockIdx.x * 16 + m16;   // A row for this lane
    const int koff  = hi * 2;                  // K offset within the 4-wide step
    const int col   = wave * 16 + m16;         // B/C column for this lane

    const float* arow = A + (size_t)row * K + koff;

    v8f acc = {};
#pragma unroll 4
    for (int k = 0; k < K; k += 4) {
        v2f a = *(const v2f*)(arow + k);       // 8B-aligned: K even, koff even
        v2f b;
        b.x = B[(size_t)(k + koff)     * N + col];
        b.y = B[(size_t)(k + koff + 1) * N + col];
        acc = __builtin_amdgcn_wmma_f32_16x16x4_f32(
            /*neg_a=*/false, a, /*neg_b=*/false, b,
            /*c_mod=*/(short)0, acc, /*reuse_a=*/false, /*reuse_b=*/false);
    }

    const int crow = blockIdx.x * 16 + hi * 8;
#pragma unroll
    for (int r = 0; r < 8; ++r)
        C[(size_t)(crow + r) * N + col] = acc[r];
}

// ---------------------------------------------------------------------------
// Edge scatter: agg[dst] += h[src] * dinv[src]*dinv[dst]
// thread-per-(edge,feature); hardware f32 atomics (L2-resident working set)
// ---------------------------------------------------------------------------
template <int F, int LOGF>
__global__ void scatter_kernel(const float* __restrict__ h,
                               const int*   __restrict__ src,
                               const int*   __restrict__ dst,
                               const float* __restrict__ dinv,
                               float*       __restrict__ agg,
                               int nE) {
    int idx = blockIdx.x * blockDim.x + threadIdx.x;   // nE*F < 2^31
    if (idx >= nE * F) return;
    const int e = idx >> LOGF;
    const int f = idx & (F - 1);
    const int s = src[e];
    const int d = dst[e];
    const float c = dinv[s] * dinv[d];
    unsafeAtomicAdd(&agg[(size_t)d * F + f], h[(size_t)s * F + f] * c);
}

// ---------------------------------------------------------------------------
// out = relu(agg + h * dinv^2 + bias)     (out may alias agg: elementwise)
// ---------------------------------------------------------------------------
template <int F, int LOGF>
__global__ void finalize_kernel(const float* __restrict__ agg,
                                const float* __restrict__ h,
                                const float* __restrict__ dinv,
                                const float* __restrict__ bias,
                                float*       __restrict__ out,
                                int nNodes) {
    int idx = blockIdx.x * blockDim.x + threadIdx.x;
    if (idx >= nNodes * F) return;
    const int i = idx >> LOGF;
    const int f = idx & (F - 1);
    const float di = dinv[i];
    const float v = agg[idx] + h[idx] * (di * di) + bias[f];
    out[idx] = v > 0.0f ? v : 0.0f;
}

// ---------------------------------------------------------------------------
extern "C" void kernel_launch(void* const* d_in, const int* in_sizes, int n_in,
                              void* d_out, int out_size, void* d_ws, size_t ws_size,
                              hipStream_t stream) {
    const float* x   = (const float*)d_in[0];   // [N, 256]
    const int*   ei  = (const int*)  d_in[1];   // [2, E] row-major
    const float* W1  = (const float*)d_in[2];   // [256, 128]
    const float* b1  = (const float*)d_in[3];   // [128]
    const float* W2  = (const float*)d_in[4];   // [128, 64]
    const float* b2  = (const float*)d_in[5];   // [64]
    float* out = (float*)d_out;                 // [N, 64]

    const int nE = in_sizes[1] / 2;
    const int* src = ei;
    const int* dst = ei + nE;

    // workspace layout (all f32):
    //   dinv : N                  (doubles as int degree counts first)
    //   bufA : N*128              (h1; later h2 in [0,N*64) and agg2 in [N*64, N*128))
    //   bufB : N*128              (agg1; finalize1 makes it h in-place)
    char* w = (char*)d_ws;
    float* dinv = (float*)w;                     w += ((size_t)N_NODES * 4 + 255) & ~(size_t)255;
    float* bufA = (float*)w;                     w += (size_t)N_NODES * HIDDEN * 4;
    float* bufB = (float*)w;
    float* h2   = bufA;
    float* agg2 = bufA + (size_t)N_NODES * OUT_DIM;

    const int B = 256;

    // --- degree / normalization -------------------------------------------
    zero_i32_kernel<<<cdiv(N_NODES, B), B, 0, stream>>>((int*)dinv, N_NODES);
    count_deg_kernel<<<cdiv(nE, B), B, 0, stream>>>(dst, (int*)dinv, nE);
    make_dinv_kernel<<<cdiv(N_NODES, B), B, 0, stream>>>(dinv, N_NODES);

    // --- layer 1 -----------------------------------------------------------
    // h1 = x @ W1   (6250 blocks x 8 waves, v_wmma_f32_16x16x4_f32)
    gemm_wmma_kernel<IN_DIM, HIDDEN>
        <<<N_NODES / 16, 32 * (HIDDEN / 16), 0, stream>>>(x, W1, bufA);

    zero_f32_kernel<<<cdiv((long long)N_NODES * HIDDEN, B), B, 0, stream>>>(
        bufB, N_NODES * HIDDEN);
    scatter_kernel<HIDDEN, 7><<<cdiv((long long)nE * HIDDEN, B), B, 0, stream>>>(
        bufA, src, dst, dinv, bufB, nE);
    finalize_kernel<HIDDEN, 7><<<cdiv((long long)N_NODES * HIDDEN, B), B, 0, stream>>>(
        bufB, bufA, dinv, b1, bufB, N_NODES);   // in-place: bufB becomes h

    // --- layer 2 -----------------------------------------------------------
    // h2 = h @ W2   (6250 blocks x 4 waves)
    gemm_wmma_kernel<HIDDEN, OUT_DIM>
        <<<N_NODES / 16, 32 * (OUT_DIM / 16), 0, stream>>>(bufB, W2, h2);

    zero_f32_kernel<<<cdiv((long long)N_NODES * OUT_DIM, B), B, 0, stream>>>(
        agg2, N_NODES * OUT_DIM);
    scatter_kernel<OUT_DIM, 6><<<cdiv((long long)nE * OUT_DIM, B), B, 0, stream>>>(
        h2, src, dst, dinv, agg2, nE);
    finalize_kernel<OUT_DIM, 6><<<cdiv((long long)N_NODES * OUT_DIM, B), B, 0, stream>>>(
        agg2, h2, dinv, b2, out, N_NODES);
}